// GraphEncoder_7361573945538
// MI455X (gfx1250) — compile-verified
//
#include <hip/hip_runtime.h>
#include <math.h>

#define HDIM 128           // hidden dim (fixed by the model)
#define CPL  4             // channels per lane (128 / wave32)
#define NEGS 0.2f          // LeakyReLU slope
#define BNEPS 1e-5f

typedef float v2f __attribute__((ext_vector_type(2)));
typedef float v8f __attribute__((ext_vector_type(8)));

// ---------------------------------------------------------------------------
// WMMA f32 GEMM:  C[M x P] = A[M x K] @ B[K x P]   (all row-major, fp32)
// One wave32 computes one 16x16 tile of C using V_WMMA_F32_16X16X4_F32.
// Requires M%16==0, K%4==0, P%16==0 (true for all GEMMs in this model).
// Fragment layout per CDNA5 ISA 7.12.2 (h = lane>>4, c = lane&15):
//   A 16x4 : VGPR i = A[row0+c][k + 2h + i]
//   B 4x16 : VGPR i = B[k + 2h + i][col0 + c]
//   C 16x16: VGPR i = C[row0 + i + 8h][col0 + c]
// ---------------------------------------------------------------------------
__global__ __launch_bounds__(256)
void wmma_gemm_f32(const float* __restrict__ A, const float* __restrict__ B,
                   float* __restrict__ C, int M, int K, int P)
{
    const int wavesPerBlock = blockDim.x >> 5;
    const int wid    = blockIdx.x * wavesPerBlock + (threadIdx.x >> 5);
    const int tilesP = P >> 4;
    const int tM     = wid / tilesP;
    const int tP     = wid - tM * tilesP;
    if (tM * 16 >= M) return;                 // wave-uniform exit (EXEC stays all-1)
    const int lane = threadIdx.x & 31;
    const int h    = lane >> 4;
    const int c    = lane & 15;
    const int row0 = tM << 4;
    const int col0 = tP << 4;

    v8f acc = {0.f, 0.f, 0.f, 0.f, 0.f, 0.f, 0.f, 0.f};
    for (int k = 0; k < K; k += 4) {
        const float* ap = A + (size_t)(row0 + c) * K + (k + 2 * h);
        v2f a; a.x = ap[0]; a.y = ap[1];
        const float* bp = B + (size_t)(k + 2 * h) * P + col0 + c;
        v2f b; b.x = bp[0]; b.y = bp[P];
        acc = __builtin_amdgcn_wmma_f32_16x16x4_f32(
                  /*neg_a=*/false, a, /*neg_b=*/false, b,
                  /*c_mod=*/(short)0, acc, /*reuse_a=*/false, /*reuse_b=*/false);
    }
    float* cp = C + (size_t)(row0 + 8 * h) * P + col0 + c;
    #pragma unroll
    for (int i = 0; i < 8; ++i) cp[(size_t)i * P] = acc[i];
}

// ---------------------------------------------------------------------------
// Helpers
// ---------------------------------------------------------------------------
__global__ void fill_f32(float* __restrict__ p, float v, int n)
{
    int i = blockIdx.x * blockDim.x + threadIdx.x;
    if (i < n) p[i] = v;
}

__global__ void init_bias(float* __restrict__ out, const float* __restrict__ b, int total)
{
    int i = blockIdx.x * blockDim.x + threadIdx.x;
    if (i < total) out[i] = b[i & (HDIM - 1)];
}

__global__ void elu_inplace(float* __restrict__ h, int total)
{
    int i = blockIdx.x * blockDim.x + threadIdx.x;
    if (i >= total) return;
    float v = h[i];
    h[i] = v > 0.f ? v : (__expf(v) - 1.f);
}

__device__ inline void atomicMaxFloat(float* addr, float v)
{
    // Race-safe ordered-int trick; addr must be pre-initialized to -inf.
    if (v >= 0.f) atomicMax((int*)addr, __float_as_int(v));
    else          atomicMin((unsigned int*)addr, __float_as_uint(v));
}

// ---------------------------------------------------------------------------
// Edge pass 1: one wave per edge.
//   m[c] = XL[src][c] + XR[dst][c] (+ sum_k ea[k]*We[k][c])
//   a    = dot(leakyrelu(m), att);   amax[dst] = max(...)
// ---------------------------------------------------------------------------
__global__ __launch_bounds__(256)
void edge_logits(const float* __restrict__ XL, const float* __restrict__ XR,
                 const float* __restrict__ EA, const float* __restrict__ We,
                 const int* __restrict__ src, const int* __restrict__ dst,
                 const float* __restrict__ att,
                 float* __restrict__ a_out, float* __restrict__ amax,
                 int Etot, int DEd)
{
    const int e    = (blockIdx.x * blockDim.x + threadIdx.x) >> 5;
    const int lane = threadIdx.x & 31;
    if (e >= Etot) return;                    // wave-uniform
    const int s = src[e], d = dst[e];
    const float* xl = XL + (size_t)s * HDIM + lane * CPL;
    const float* xr = XR + (size_t)d * HDIM + lane * CPL;

    float m[CPL];
    #pragma unroll
    for (int i = 0; i < CPL; ++i) m[i] = xl[i] + xr[i];

    if (EA != nullptr) {
        const float* ea = EA + (size_t)e * DEd;
        for (int k = 0; k < DEd; ++k) {
            float ev = ea[k];                 // broadcast load (L0/L2 hit)
            const float* w = We + (size_t)k * HDIM + lane * CPL;
            #pragma unroll
            for (int i = 0; i < CPL; ++i) m[i] += ev * w[i];
        }
    }

    float part = 0.f;
    #pragma unroll
    for (int i = 0; i < CPL; ++i) {
        float v = m[i];
        v = v > 0.f ? v : NEGS * v;           // LeakyReLU(0.2)
        part += v * att[lane * CPL + i];
    }
    // wave32 reduction
    for (int off = 16; off > 0; off >>= 1) part += __shfl_down(part, off, 32);

    if (lane == 0) {
        a_out[e] = part;
        atomicMaxFloat(&amax[d], part);
    }
}

// Edge pass 2: one thread per edge.  ex = exp(a - amax[dst]);  den[dst] += ex
__global__ void edge_expden(const float* __restrict__ a, const int* __restrict__ dst,
                            const float* __restrict__ amax,
                            float* __restrict__ ex, float* __restrict__ den, int Etot)
{
    int e = blockIdx.x * blockDim.x + threadIdx.x;
    if (e >= Etot) return;
    int d = dst[e];
    float v = __expf(a[e] - amax[d]);
    ex[e] = v;
    atomicAdd(&den[d], v);
}

// Edge pass 3: one wave per edge.  out[dst] += (ex/den[dst]) * XL[src]
__global__ __launch_bounds__(256)
void edge_aggregate(const float* __restrict__ XL, const float* __restrict__ ex,
                    const float* __restrict__ den,
                    const int* __restrict__ src, const int* __restrict__ dst,
                    float* __restrict__ out, int Etot)
{
    const int e    = (blockIdx.x * blockDim.x + threadIdx.x) >> 5;
    const int lane = threadIdx.x & 31;
    if (e >= Etot) return;                    // wave-uniform
    const int s = src[e], d = dst[e];
    const float alpha = ex[e] / den[d];
    const float* xl = XL + (size_t)s * HDIM + lane * CPL;
    float* o        = out + (size_t)d * HDIM + lane * CPL;
    #pragma unroll
    for (int i = 0; i < CPL; ++i) atomicAdd(&o[i], alpha * xl[i]);
}

// ---------------------------------------------------------------------------
// BatchNorm (training-mode, biased var)
// ---------------------------------------------------------------------------
__global__ __launch_bounds__(HDIM)
void bn_stats(const float* __restrict__ h, float* __restrict__ sum,
              float* __restrict__ sq, int Nn)
{
    const int c = threadIdx.x;                // one channel per thread
    float s = 0.f, q = 0.f;
    for (int r = blockIdx.x; r < Nn; r += gridDim.x) {
        float v = h[(size_t)r * HDIM + c];    // block reads full rows, coalesced
        s += v; q += v * v;
    }
    atomicAdd(&sum[c], s);
    atomicAdd(&sq[c], q);
}

__global__ void bn_finalize(const float* __restrict__ sum, const float* __restrict__ sq,
                            float* __restrict__ mu, float* __restrict__ inv, int Nn)
{
    int c = threadIdx.x;
    float m = sum[c] / (float)Nn;
    float v = sq[c] / (float)Nn - m * m;
    mu[c]  = m;
    inv[c] = rsqrtf(v + BNEPS);
}

__global__ void bn_apply(float* __restrict__ h, const float* __restrict__ mu,
                         const float* __restrict__ inv, const float* __restrict__ g,
                         const float* __restrict__ be, int total, int eluAfter)
{
    int i = blockIdx.x * blockDim.x + threadIdx.x;
    if (i >= total) return;
    int c = i & (HDIM - 1);
    float v = (h[i] - mu[c]) * inv[c] * g[c] + be[c];
    if (eluAfter) v = v > 0.f ? v : (__expf(v) - 1.f);
    h[i] = v;
}

// ---------------------------------------------------------------------------
// Launch
// ---------------------------------------------------------------------------
extern "C" void kernel_launch(void* const* d_in, const int* in_sizes, int n_in,
                              void* d_out, int out_size, void* d_ws, size_t ws_size,
                              hipStream_t stream)
{
    const float* x    = (const float*)d_in[0];
    const int*   ei   = (const int*)  d_in[1];
    const float* ea   = (const float*)d_in[2];
    const float* Wl1  = (const float*)d_in[3];
    const float* Wr1  = (const float*)d_in[4];
    const float* We1  = (const float*)d_in[5];
    const float* att1 = (const float*)d_in[6];
    const float* b1   = (const float*)d_in[7];
    const float* Wl2  = (const float*)d_in[8];
    const float* Wr2  = (const float*)d_in[9];
    const float* att2 = (const float*)d_in[10];
    const float* b2   = (const float*)d_in[11];
    const float* Wl3  = (const float*)d_in[12];
    const float* Wr3  = (const float*)d_in[13];
    const float* att3 = (const float*)d_in[14];
    const float* b3   = (const float*)d_in[15];
    const float* g0   = (const float*)d_in[16];
    const float* be0  = (const float*)d_in[17];
    const float* g1   = (const float*)d_in[18];
    const float* be1  = (const float*)d_in[19];

    const int Etot = in_sizes[1] / 2;           // E + N (self loops)
    const int DIN  = in_sizes[3] / HDIM;        // 64
    const int Nn   = in_sizes[0] / DIN;         // 50000
    const int DEd  = in_sizes[5] / HDIM;        // 16
    const int* srcI = ei;
    const int* dstI = ei + Etot;

    // ---- workspace layout (floats) ----
    float* ws = (float*)d_ws;
    float* XL  = ws;  ws += (size_t)Nn * HDIM;
    float* XR  = ws;  ws += (size_t)Nn * HDIM;
    float* Hb  = ws;  ws += (size_t)Nn * HDIM;  // inter-layer features
    float* Ae  = ws;  ws += Etot;               // logits
    float* Ex  = ws;  ws += Etot;               // exp(logit - max)
    float* Am  = ws;  ws += Nn;                 // segment max
    float* Dn  = ws;  ws += Nn;                 // segment sum (softmax den)
    float* Ssum = ws; ws += HDIM;               // BN sums (Ssum,Ssq contiguous)
    float* Ssq  = ws; ws += HDIM;
    float* Mu   = ws; ws += HDIM;
    float* Inv  = ws; ws += HDIM;

    const int ntot    = Nn * HDIM;
    const int nblocks = (ntot + 255) / 256;
    const int eWblk   = (Etot + 7) / 8;         // 8 waves/block, 1 wave/edge
    const int eTblk   = (Etot + 255) / 256;
    const int nFill   = (Nn + 255) / 256;

    auto gemm = [&](const float* A, const float* B, float* C, int M, int K, int P) {
        int waves  = (M / 16) * (P / 16);
        int blocks = (waves + 7) / 8;
        wmma_gemm_f32<<<blocks, 256, 0, stream>>>(A, B, C, M, K, P);
    };

    auto run_layer = [&](const float* featIn, int Kdim,
                         const float* Wl, const float* Wr,
                         const float* attv, const float* bias,
                         const float* EAp, const float* Wep,
                         float* outFeat) {
        gemm(featIn, Wl, XL, Nn, Kdim, HDIM);
        gemm(featIn, Wr, XR, Nn, Kdim, HDIM);
        fill_f32<<<nFill, 256, 0, stream>>>(Am, -INFINITY, Nn);
        fill_f32<<<nFill, 256, 0, stream>>>(Dn, 0.f, Nn);
        edge_logits<<<eWblk, 256, 0, stream>>>(XL, XR, EAp, Wep, srcI, dstI,
                                               attv, Ae, Am, Etot, DEd);
        edge_expden<<<eTblk, 256, 0, stream>>>(Ae, dstI, Am, Ex, Dn, Etot);
        init_bias<<<nblocks, 256, 0, stream>>>(outFeat, bias, ntot);  // out = b
        edge_aggregate<<<eWblk, 256, 0, stream>>>(XL, Ex, Dn, srcI, dstI,
                                                  outFeat, Etot);     // out += segsum
    };

    auto batchnorm = [&](float* feat, const float* g, const float* be, int eluAfter) {
        fill_f32<<<1, 256, 0, stream>>>(Ssum, 0.f, 2 * HDIM);  // zeros Ssum+Ssq
        bn_stats<<<512, HDIM, 0, stream>>>(feat, Ssum, Ssq, Nn);
        bn_finalize<<<1, HDIM, 0, stream>>>(Ssum, Ssq, Mu, Inv, Nn);
        bn_apply<<<nblocks, 256, 0, stream>>>(feat, Mu, Inv, g, be, ntot, eluAfter);
    };

    // Layer 1: GATv2(+edge_attr) -> ELU -> BN
    run_layer(x, DIN, Wl1, Wr1, att1, b1, ea, We1, Hb);
    elu_inplace<<<nblocks, 256, 0, stream>>>(Hb, ntot);
    batchnorm(Hb, g0, be0, /*eluAfter=*/0);

    // Layer 2: GATv2 -> BN -> ELU
    run_layer(Hb, HDIM, Wl2, Wr2, att2, b2, nullptr, nullptr, Hb);
    batchnorm(Hb, g1, be1, /*eluAfter=*/1);

    // Layer 3: GATv2 -> output
    run_layer(Hb, HDIM, Wl3, Wr3, att3, b3, nullptr, nullptr, (float*)d_out);
}